// GCNSparse_69604239999348
// MI455X (gfx1250) — compile-verified
//
#include <hip/hip_runtime.h>

typedef __attribute__((ext_vector_type(2))) float v2f;
typedef __attribute__((ext_vector_type(8))) float v8f;

// ---------------------------------------------------------------------------
// Dense GEMM, row-major: C[M,Nc] = A[M,K] @ B[K,Nc].
// One wave32 computes a 16 x (16*NT) strip of C via V_WMMA_F32_16X16X4_F32:
// NT accumulators share each A fragment (4x less A traffic, 4 WMMAs per A
// load). K stepped by 4. Dims are multiples of 16 here (100000/256/64).
// ---------------------------------------------------------------------------
template <int NT>
__global__ void __launch_bounds__(256)
gcn_wmma_gemm_f32(const float* __restrict__ A, const float* __restrict__ B,
                  float* __restrict__ C, int M, int K, int Nc) {
  const int wavesPerBlock = blockDim.x >> 5;
  const int waveId = blockIdx.x * wavesPerBlock + ((int)threadIdx.x >> 5);
  const int groupsPerRow = Nc / (16 * NT);
  const int tileM = waveId / groupsPerRow;
  const int group = waveId - tileM * groupsPerRow;
  if (tileM * 16 >= M) return;              // wave-uniform: EXEC stays all-1s

  const int lane = threadIdx.x & 31;
  const int half = lane >> 4;               // 0: lanes 0-15, 1: lanes 16-31
  const int l    = lane & 15;
  const int n0   = group * (16 * NT);

  // A 16x4 frag (ISA 7.12.2): lanes 0-15 hold A[m=l][k0+0],[k0+1] in v0,v1;
  // lanes 16-31 hold A[m=l][k0+2],[k0+3].
  const float* ap = A + (size_t)(tileM * 16 + l) * K + 2 * half;
  // B 4x16 frags (symmetric over N): lanes 0-15 hold B[k0+0][n],B[k0+1][n];
  // lanes 16-31 hold B[k0+2][n],B[k0+3][n].
  const float* bp = B + (size_t)(2 * half) * Nc + n0 + l;

  v8f acc[NT];
#pragma unroll
  for (int i = 0; i < NT; ++i) acc[i] = (v8f){0.f,0.f,0.f,0.f,0.f,0.f,0.f,0.f};

#pragma unroll 2
  for (int k0 = 0; k0 < K; k0 += 4) {
    v2f a;
    a[0] = ap[0];
    a[1] = ap[1];
#pragma unroll
    for (int i = 0; i < NT; ++i) {
      v2f b;
      b[0] = bp[i * 16];
      b[1] = bp[i * 16 + Nc];
      // 8 args: (neg_a, A, neg_b, B, c_mod, C, reuse_a, reuse_b)
      acc[i] = __builtin_amdgcn_wmma_f32_16x16x4_f32(
          false, a, false, b, (short)0, acc[i], false, false);
    }
    ap += 4;
    bp += (size_t)4 * Nc;
  }

  // D layout: VGPR r, lanes 0-15 -> C[tileM*16 + r][n], lanes 16-31 -> row +8.
  float* cp = C + (size_t)(tileM * 16 + half * 8) * Nc + n0 + l;
#pragma unroll
  for (int i = 0; i < NT; ++i) {
#pragma unroll
    for (int r = 0; r < 8; ++r) cp[(size_t)r * Nc + i * 16] = acc[i][r];
  }
}

// ---------------------------------------------------------------------------
// Deterministic SpMM: out[r, :] = sum_{e: row[e]==r} ev[e] * dense[col[e], :]
//                     (+ bias, optional relu).
// `row` is sorted, so each block binary-searches its row's [start,end) edge
// range and accumulates in a fixed order -> bit-identical across replays, no
// atomics (graph-capture safe). blockDim.x == F; thread t owns feature t.
// 4-wide unroll gives 4 independent gathers in flight per wave; ev/col are
// wave-uniform -> scalar loads. dense (102MB / 25MB) is L2-resident (192MB).
// ---------------------------------------------------------------------------
template <int F, bool RELU>
__global__ void __launch_bounds__(F)
gcn_spmm_rows(const float* __restrict__ dense,
              const float* __restrict__ ev,
              const int* __restrict__ row,
              const int* __restrict__ col,
              const float* __restrict__ bias,
              float* __restrict__ out,
              int nEdges) {
  const int r = blockIdx.x;

  // lower_bound(row, r)
  int lo = 0, hi = nEdges;
  while (lo < hi) {
    int mid = (lo + hi) >> 1;
    if (row[mid] < r) lo = mid + 1; else hi = mid;
  }
  const int start = lo;
  // lower_bound(row, r+1), starting from `start`
  hi = nEdges;
  while (lo < hi) {
    int mid = (lo + hi) >> 1;
    if (row[mid] <= r) lo = mid + 1; else hi = mid;
  }
  const int end = lo;

  const int t = threadIdx.x;
  float acc = 0.f;
  int e = start;
  for (; e + 3 < end; e += 4) {
    float v0 = ev[e + 0], v1 = ev[e + 1], v2 = ev[e + 2], v3 = ev[e + 3];
    int   c0 = col[e + 0], c1 = col[e + 1], c2 = col[e + 2], c3 = col[e + 3];
    float d0 = dense[(size_t)c0 * F + t];
    float d1 = dense[(size_t)c1 * F + t];
    float d2 = dense[(size_t)c2 * F + t];
    float d3 = dense[(size_t)c3 * F + t];
    acc += v0 * d0;
    acc += v1 * d1;
    acc += v2 * d2;
    acc += v3 * d3;
  }
  for (; e < end; ++e) acc += ev[e] * dense[(size_t)col[e] * F + t];

  acc += bias[t];
  if (RELU) acc = fmaxf(acc, 0.f);
  out[(size_t)r * F + t] = acc;
}

// ---------------------------------------------------------------------------
// Inputs (setup_inputs order): X, edge_vals, W1, b1, W2, b2, row, col
// ---------------------------------------------------------------------------
extern "C" void kernel_launch(void* const* d_in, const int* in_sizes, int n_in,
                              void* d_out, int out_size, void* d_ws, size_t ws_size,
                              hipStream_t stream) {
  const float* X  = (const float*)d_in[0];
  const float* ev = (const float*)d_in[1];
  const float* W1 = (const float*)d_in[2];
  const float* b1 = (const float*)d_in[3];
  const float* W2 = (const float*)d_in[4];
  const float* b2 = (const float*)d_in[5];
  const int*   row = (const int*)d_in[6];
  const int*   col = (const int*)d_in[7];
  float* out = (float*)d_out;

  const int IN  = 256;
  const int HID = 256;
  const int OUT = 64;
  const int N   = in_sizes[0] / IN;   // 100000
  const int E   = in_sizes[1];        // 3200000

  // Workspace layout: H0 [N*HID f32] | Hs [N*HID f32]; H2 reuses H0's region.
  float* H0 = (float*)d_ws;                  // X @ W1
  float* Hs = H0 + (size_t)N * HID;          // relu(spmm(H0) + b1)
  float* H2 = H0;                            // Hs @ W2 (reuse, H0 dead by then)

  const int NT = 4;                          // 16x64 strip per wave

  // 1) H0 = X @ W1   (100000x256 @ 256x256)
  {
    int waves  = (N / 16) * (HID / (16 * NT));   // 6250 * 4 = 25000
    int blocks = (waves + 7) / 8;                // 8 waves / 256-thread block
    gcn_wmma_gemm_f32<NT><<<blocks, 256, 0, stream>>>(X, W1, H0, N, IN, HID);
  }

  // 2) Hs = relu(A @ H0 + b1)
  gcn_spmm_rows<256, true><<<N, 256, 0, stream>>>(H0, ev, row, col, b1, Hs, E);

  // 3) H2 = Hs @ W2  (100000x256 @ 256x64)
  {
    int waves  = (N / 16) * (OUT / (16 * NT));   // 6250 * 1 = 6250
    int blocks = (waves + 7) / 8;                // 782 blocks (tail returns)
    gcn_wmma_gemm_f32<NT><<<blocks, 256, 0, stream>>>(Hs, W2, H2, N, HID, OUT);
  }

  // 4) out = A @ H2 + b2
  gcn_spmm_rows<64, false><<<N, 64, 0, stream>>>(H2, ev, row, col, b2, out, E);
}